// GaussianProcessLayer_2267742732495
// MI455X (gfx1250) — compile-verified
//
#include <hip/hip_runtime.h>

// GP layer for MI455X (gfx1250, wave32). All GEMM-shaped math goes through
// v_wmma_f32_16x16x4_f32 (FP32 matrix pipe); the covariance builder stages its
// block-shared panels into LDS with the Tensor Data Mover (tensor_load_to_lds
// + s_wait_tensorcnt). Choleskys / trisolves are scalar.

#define UNITS 128
#define M     64
#define N     512
#define DD    128
#define JITTER 1e-6f
#define PNV    0.1f
#define UCHUNK 16   // cov slab: 16 units * 1MB = 16MB, L2-resident

typedef float v2f __attribute__((ext_vector_type(2)));
typedef float v8f __attribute__((ext_vector_type(8)));
typedef unsigned int v4u __attribute__((ext_vector_type(4)));
typedef int v4i __attribute__((ext_vector_type(4)));
typedef int v8i __attribute__((ext_vector_type(8)));

__device__ __forceinline__ v8f wmma_f32_4(v2f a, v2f b, v8f c) {
  // D = A(16x4) * B(4x16) + C, fp32
  return __builtin_amdgcn_wmma_f32_16x16x4_f32(false, a, false, b, (short)0, c,
                                               false, false);
}

// ---- TDM: 2-D tile (tile0 x tile1 elems of 4B) global -> LDS -------------
// D# built per CDNA5 ISA 8.3/8.4: group0 = {count/type/addr/lds},
// group1 = {data_size, dims, tile dims, dim0 stride}.
#if defined(__gfx1250__) && __has_builtin(__builtin_amdgcn_tensor_load_to_lds)
#define HAVE_TDM 1
__device__ __forceinline__ void tdm_load_2d_f32(const float* gsrc,
                                                unsigned ldsByteOff,
                                                unsigned tdim0, unsigned tdim1,
                                                unsigned tile0, unsigned tile1,
                                                unsigned stride0) {
  unsigned long long ga = (unsigned long long)(uintptr_t)gsrc;
  v4u g0 = {1u,                                    // count=1, is_restore=0
            ldsByteOff,                            // lds_addr
            (unsigned)(ga & 0xFFFFFFFFu),          // global_addr[31:0]
            (unsigned)((ga >> 32) & 0x01FFFFFFu) | (2u << 30)}; // [56:32]|type=2
  v8i g1 = {(int)0x00020000u,                                   // data_size=4B
            (int)((tdim0 & 0xFFFFu) << 16),                     // dim0 lo
            (int)(((tdim0 >> 16) & 0xFFFFu) | ((tdim1 & 0xFFFFu) << 16)),
            (int)(((tdim1 >> 16) & 0xFFFFu) | ((tile0 & 0xFFFFu) << 16)),
            (int)(tile1 & 0xFFFFu),                             // tile1, tile2=0
            (int)stride0,                                       // dim0 stride lo
            0, 0};
  v4i z4 = {0, 0, 0, 0};
#if __clang_major__ >= 23
  v8i z8 = {0, 0, 0, 0, 0, 0, 0, 0};
  __builtin_amdgcn_tensor_load_to_lds(g0, g1, z4, z4, z8, 0);
#else
  __builtin_amdgcn_tensor_load_to_lds(g0, g1, z4, z4, 0);
#endif
}
__device__ __forceinline__ void tdm_wait() {
#if __has_builtin(__builtin_amdgcn_s_wait_tensorcnt)
  __builtin_amdgcn_s_wait_tensorcnt(0);
#else
  asm volatile("s_wait_tensorcnt 0x0" ::: "memory");
#endif
}
#else
#define HAVE_TDM 0
#endif

// ---------------- row squared-norms -------------------------------------
__global__ void row_norms_kernel(const float* __restrict__ P,
                                 float* __restrict__ out, int rows) {
  int r = blockIdx.x * blockDim.x + threadIdx.x;
  if (r >= rows) return;
  const float4* p = (const float4*)(P + (size_t)r * DD);
  float s = 0.f;
#pragma unroll
  for (int d = 0; d < DD / 4; ++d) {
    float4 v = p[d];
    s += v.x * v.x + v.y * v.y + v.z * v.z + v.w * v.w;
  }
  out[r] = s;
}

// ---------------- RBF gram via WMMA -------------------------------------
__global__ void rbf_wmma_kernel(const float* __restrict__ A,
                                const float* __restrict__ B,
                                const float* __restrict__ nA,
                                const float* __restrict__ nB,
                                float* __restrict__ Out,
                                size_t sAU, size_t sBU, size_t snAU, size_t snBU,
                                size_t sOutU, int ldOut, int tilesX,
                                const float* __restrict__ ampP,
                                const float* __restrict__ lsP, float diagAdd) {
  int lane = threadIdx.x & 31, wid = threadIdx.x >> 5;
  int tx = blockIdx.x * 4 + wid;
  if (tx >= tilesX) return;                 // wave-uniform
  int half = lane >> 4, lp = lane & 15;
  int u = blockIdx.z;
  const float* Au = A + (size_t)u * sAU;
  const float* Bu = B + (size_t)u * sBU;
  int i0 = blockIdx.y * 16, j0 = tx * 16;
  const float* arow = Au + (size_t)(i0 + lp) * DD + 2 * half;
  const float* brow = Bu + (size_t)(j0 + lp) * DD + 2 * half;
  v8f acc = {0.f, 0.f, 0.f, 0.f, 0.f, 0.f, 0.f, 0.f};
#pragma unroll
  for (int k = 0; k < DD; k += 4) {
    v2f a, b;
    a.x = arow[k]; a.y = arow[k + 1];
    b.x = brow[k]; b.y = brow[k + 1];
    acc = wmma_f32_4(a, b, acc);
  }
  float amp = ampP[0], ls = lsP[0];
  float amp2 = amp * amp, inv_ls2 = 1.0f / (ls * ls);
  const float* nAu = nA + (size_t)u * snAU;
  const float* nBu = nB + (size_t)u * snBU;
  float nb = nBu[j0 + lp];
  float* outU = Out + (size_t)u * sOutU;
#pragma unroll
  for (int r = 0; r < 8; ++r) {
    int i = i0 + r + 8 * half, j = j0 + lp;
    float d2 = (nAu[i] + nb - 2.0f * acc[r]) * inv_ls2;
    d2 = fmaxf(d2, 0.0f);
    float v = amp2 * __expf(-0.5f * d2);
    if (i == j) v += diagAdd;
    outU[(size_t)i * ldOut + j] = v;
  }
}

// ---------------- 64x64 Cholesky (one block/unit, 64 threads) -----------
__global__ void cholesky64_kernel(float* __restrict__ Kmm) {
  __shared__ float T[64 * 65];
  int u = blockIdx.x, tid = threadIdx.x;
  float* Ku = Kmm + (size_t)u * (M * M);
  for (int idx = tid; idx < M * M; idx += 64)
    T[(idx >> 6) * 65 + (idx & 63)] = Ku[idx];
  __syncthreads();
  for (int k = 0; k < M; ++k) {
    if (tid == k) T[k * 65 + k] = sqrtf(T[k * 65 + k]);
    __syncthreads();
    float lkk = T[k * 65 + k];
    if (tid > k) T[tid * 65 + k] /= lkk;
    __syncthreads();
    float lik = (tid > k) ? T[tid * 65 + k] : 0.0f;
    for (int j = k + 1; j <= tid; ++j)
      T[tid * 65 + j] -= lik * T[j * 65 + k];
    __syncthreads();
  }
  for (int j = tid + 1; j < M; ++j) T[tid * 65 + j] = 0.0f;
  __syncthreads();
  for (int idx = tid; idx < M * M; idx += 64)
    Ku[idx] = T[(idx >> 6) * 65 + (idx & 63)];
}

// ---------------- forward substitution: A=L^-1 Kmn (in place), C, alpha --
__global__ void trisolve_kernel(const float* __restrict__ L,
                                float* __restrict__ Akmn,
                                const float* __restrict__ S,
                                float* __restrict__ C,
                                const float* __restrict__ mloc,
                                float* __restrict__ alpha) {
  __shared__ float Ls[64 * 65];
  int u = blockIdx.x, tid = threadIdx.x;   // blockDim = 640
  const float* Lu = L + (size_t)u * (M * M);
  for (int idx = tid; idx < M * M; idx += blockDim.x)
    Ls[(idx >> 6) * 65 + (idx & 63)] = Lu[idx];
  __syncthreads();
  const float* rp; float* op; int stride;
  if (tid < N) {
    op = Akmn + (size_t)u * M * N + tid; rp = op; stride = N;
  } else if (tid < N + M) {
    int j = tid - N;
    rp = S + (size_t)u * M * M + j; op = C + (size_t)u * M * M + j; stride = M;
  } else if (tid == N + M) {
    rp = mloc + (size_t)u * M; op = alpha + (size_t)u * M; stride = 1;
  } else {
    return;
  }
  float a[M];
#pragma unroll
  for (int i = 0; i < M; ++i) {
    float t = rp[(size_t)i * stride];
#pragma unroll
    for (int j = 0; j < i; ++j) t -= Ls[i * 65 + j] * a[j];
    a[i] = t / Ls[i * 65 + i];
    op[(size_t)i * stride] = a[i];
  }
}

// ---------------- mu[u][n] = sum_m A[u][m][n] * alpha[u][m] --------------
__global__ void mu_kernel(const float* __restrict__ A,
                          const float* __restrict__ alpha,
                          float* __restrict__ mu) {
  int u = blockIdx.x, n = threadIdx.x;     // blockDim = 512
  const float* Au = A + (size_t)u * M * N + n;
  const float* al = alpha + (size_t)u * M;
  float s = 0.f;
#pragma unroll
  for (int m = 0; m < M; ++m) s += Au[(size_t)m * N] * al[m];
  mu[(size_t)u * N + n] = s;
}

// ---------------- CA[u][m][n] = sum_j C[u][j][m] * A[u][j][n] (WMMA) ----
__global__ void ca_wmma_kernel(const float* __restrict__ C,
                               const float* __restrict__ A,
                               float* __restrict__ CA) {
  int lane = threadIdx.x & 31, wid = threadIdx.x >> 5;
  int half = lane >> 4, lp = lane & 15;
  int u = blockIdx.z;
  int n0 = (blockIdx.x * 4 + wid) * 16, m0 = blockIdx.y * 16;
  const float* Cu = C + (size_t)u * M * M;
  const float* Au = A + (size_t)u * M * N;
  v8f acc = {0.f, 0.f, 0.f, 0.f, 0.f, 0.f, 0.f, 0.f};
#pragma unroll
  for (int k = 0; k < M; k += 4) {
    int kk = k + 2 * half;
    v2f a, b;
    a.x = Cu[(size_t)kk * M + m0 + lp];
    a.y = Cu[(size_t)(kk + 1) * M + m0 + lp];
    b.x = Au[(size_t)kk * N + n0 + lp];
    b.y = Au[(size_t)(kk + 1) * N + n0 + lp];
    acc = wmma_f32_4(a, b, acc);
  }
  float* CAu = CA + (size_t)u * M * N;
#pragma unroll
  for (int r = 0; r < 8; ++r)
    CAu[(size_t)(m0 + r + 8 * half) * N + n0 + lp] = acc[r];
}

// ---------------- cov = Knn - A^T A + CA^T CA + (PNV+JIT) I --------------
// n-side panels (shared by all 4 waves of the block) staged via TDM into LDS.
__global__ void cov_wmma_kernel(const float* __restrict__ A,
                                const float* __restrict__ CA,
                                const float* __restrict__ Knn,
                                float* __restrict__ cov, int ubase) {
  __shared__ float panel[2048];  // [0..1023]=A(64x16), [1024..2047]=CA(64x16)
  int lane = threadIdx.x & 31, wid = threadIdx.x >> 5;
  int half = lane >> 4, lp = lane & 15;
  int cu = blockIdx.z, u = ubase + cu;
  int p0 = (blockIdx.x * 4 + wid) * 16, n0 = blockIdx.y * 16;
  const float* Au = A + (size_t)u * M * N;
  const float* CAu = CA + (size_t)u * M * N;

#if HAVE_TDM
  if (threadIdx.x < 32) {  // wave 0 drives the DMA; TENSORcnt is per-wave
    unsigned base = (unsigned)(unsigned long long)(uintptr_t)&panel[0];
    tdm_load_2d_f32(Au + n0, base, N, M, 16, M, N);
    tdm_load_2d_f32(CAu + n0, base + 4096u, N, M, 16, M, N);
    tdm_wait();
  }
  __syncthreads();
#else
  for (int t = threadIdx.x; t < 1024; t += blockDim.x) {
    int kk2 = t >> 4, c = t & 15;
    panel[t] = Au[(size_t)kk2 * N + n0 + c];
    panel[1024 + t] = CAu[(size_t)kk2 * N + n0 + c];
  }
  __syncthreads();
#endif

  v8f s1 = {0.f, 0.f, 0.f, 0.f, 0.f, 0.f, 0.f, 0.f};
  v8f s2 = {0.f, 0.f, 0.f, 0.f, 0.f, 0.f, 0.f, 0.f};
#pragma unroll
  for (int k = 0; k < M; k += 4) {
    int kk = k + 2 * half;
    v2f a, b;
    a.x = panel[kk * 16 + lp];              // A n-side from LDS
    a.y = panel[(kk + 1) * 16 + lp];
    b.x = Au[(size_t)kk * N + p0 + lp];     // A p-side from global
    b.y = Au[(size_t)(kk + 1) * N + p0 + lp];
    s1 = wmma_f32_4(a, b, s1);
    a.x = panel[1024 + kk * 16 + lp];       // CA n-side from LDS
    a.y = panel[1024 + (kk + 1) * 16 + lp];
    b.x = CAu[(size_t)kk * N + p0 + lp];
    b.y = CAu[(size_t)(kk + 1) * N + p0 + lp];
    s2 = wmma_f32_4(a, b, s2);
  }
  float* covU = cov + (size_t)cu * N * N;
#pragma unroll
  for (int r = 0; r < 8; ++r) {
    int n = n0 + r + 8 * half, p = p0 + lp;
    float v = Knn[(size_t)n * N + p] - s1[r] + s2[r];
    if (n == p) v += (PNV + JITTER);
    covU[(size_t)n * N + p] = v;
  }
}

// ---------------- 512x512 Cholesky (one block/unit, 512 threads) --------
__global__ void cholesky512_kernel(float* __restrict__ cov) {
  int cu = blockIdx.x, tid = threadIdx.x;
  float* Cv = cov + (size_t)cu * N * N;
  for (int k = 0; k < N; ++k) {
    if (tid == 0) Cv[(size_t)k * N + k] = sqrtf(Cv[(size_t)k * N + k]);
    if (k + 1 < N)  // pull next pivot column toward the caches
      __builtin_prefetch(&Cv[(size_t)tid * N + k + 1], 0, 1);
    __threadfence_block(); __syncthreads();
    float lkk = Cv[(size_t)k * N + k];
    if (tid > k) Cv[(size_t)tid * N + k] /= lkk;
    __threadfence_block(); __syncthreads();
    float lik = (tid > k) ? Cv[(size_t)tid * N + k] : 0.0f;
    for (int j = k + 1; j <= tid; ++j)
      Cv[(size_t)tid * N + j] -= lik * Cv[(size_t)j * N + k];
    __threadfence_block(); __syncthreads();
  }
}

// ---------------- eps: threefry2x32-20 + erfinv (key=(0,42)) ------------
__device__ __forceinline__ unsigned rotl32(unsigned x, unsigned d) {
  return (x << d) | (x >> (32u - d));
}
__device__ __forceinline__ float erfinv_f(float x) {
  float w = -__logf((1.0f - x) * (1.0f + x));
  float p;
  if (w < 5.0f) {
    w -= 2.5f;
    p = 2.81022636e-08f;
    p = fmaf(p, w, 3.43273939e-07f);  p = fmaf(p, w, -3.5233877e-06f);
    p = fmaf(p, w, -4.39150654e-06f); p = fmaf(p, w, 0.00021858087f);
    p = fmaf(p, w, -0.00125372503f);  p = fmaf(p, w, -0.00417768164f);
    p = fmaf(p, w, 0.246640727f);     p = fmaf(p, w, 1.50140941f);
  } else {
    w = sqrtf(w) - 3.0f;
    p = -0.000200214257f;
    p = fmaf(p, w, 0.000100950558f);  p = fmaf(p, w, 0.00134934322f);
    p = fmaf(p, w, -0.00367342844f);  p = fmaf(p, w, 0.00573950773f);
    p = fmaf(p, w, -0.0076224613f);   p = fmaf(p, w, 0.00943887047f);
    p = fmaf(p, w, 1.00167406f);      p = fmaf(p, w, 2.83297682f);
  }
  return p * x;
}
__global__ void eps_kernel(float* __restrict__ eps) {
  unsigned idx = blockIdx.x * blockDim.x + threadIdx.x;  // < UNITS*N
  const unsigned k0 = 0u, k1 = 42u;
  unsigned ks[3] = {k0, k1, k0 ^ k1 ^ 0x1BD11BDAu};
  unsigned x0 = 0u + ks[0], x1 = idx + ks[1];
  const int R[8] = {13, 15, 26, 6, 17, 29, 16, 24};
#pragma unroll
  for (int g = 0; g < 5; ++g) {
#pragma unroll
    for (int i = 0; i < 4; ++i) {
      x0 += x1; x1 = rotl32(x1, (unsigned)R[(g & 1) * 4 + i]); x1 ^= x0;
    }
    x0 += ks[(g + 1) % 3];
    x1 += ks[(g + 2) % 3] + (unsigned)(g + 1);
  }
  unsigned bits = x0 ^ x1;
  float u = __uint_as_float(0x3F800000u | (bits >> 9)) - 1.0f;
  float v = 2.0f * u - 1.0f;
  v = fminf(fmaxf(v, -0.9999999f), 0.9999999f);
  eps[idx] = 1.41421356237f * erfinv_f(v);
}

// ---------------- sample[u][n] = mu + sum_{p<=n} Lc[n][p] eps[p]; out.T --
__global__ void sample_kernel(const float* __restrict__ cov,
                              const float* __restrict__ mu,
                              const float* __restrict__ eps,
                              float* __restrict__ out, int ubase) {
  int cu = blockIdx.y, u = ubase + cu;
  int n = blockIdx.x * blockDim.x + threadIdx.x;
  const float* Ln = cov + (size_t)cu * N * N + (size_t)n * N;
  const float* ep = eps + (size_t)u * N;
  float s = mu[(size_t)u * N + n];
  for (int p = 0; p <= n; ++p) s += Ln[p] * ep[p];
  out[(size_t)n * UNITS + u] = s;
}

// ---------------- KL reduction ------------------------------------------
__global__ void kl_init_kernel(float* kl) {
  if (threadIdx.x == 0 && blockIdx.x == 0) kl[0] = 0.f;
}
__global__ void kl_kernel(const float* __restrict__ C,
                          const float* __restrict__ alpha,
                          const float* __restrict__ L,
                          const float* __restrict__ S,
                          float* __restrict__ kl) {
  __shared__ float red[256];
  int u = blockIdx.x, tid = threadIdx.x;
  float s = 0.f;
  const float* Cu = C + (size_t)u * M * M;
  for (int idx = tid; idx < M * M; idx += 256) { float c = Cu[idx]; s += c * c; }
  const float* al = alpha + (size_t)u * M;
  const float* Lu = L + (size_t)u * M * M;
  const float* Su = S + (size_t)u * M * M;
  for (int i = tid; i < M; i += 256) {
    s += al[i] * al[i];
    s += 2.0f * __logf(Lu[(size_t)i * M + i]);
    s -= 2.0f * __logf(fabsf(Su[(size_t)i * M + i]));
  }
  red[tid] = s; __syncthreads();
  for (int w = 128; w > 0; w >>= 1) {
    if (tid < w) red[tid] += red[tid + w];
    __syncthreads();
  }
  if (tid == 0) atomicAdd(kl, 0.5f * (red[0] - (float)M));
}

// ---------------- launch -------------------------------------------------
extern "C" void kernel_launch(void* const* d_in, const int* in_sizes, int n_in,
                              void* d_out, int out_size, void* d_ws,
                              size_t ws_size, hipStream_t stream) {
  (void)in_sizes; (void)n_in; (void)out_size; (void)ws_size;
  const float* x    = (const float*)d_in[0];  // [N, D]
  const float* z    = (const float*)d_in[1];  // [U, M, D]
  const float* loc  = (const float*)d_in[2];  // [U, M]
  const float* Svar = (const float*)d_in[3];  // [U, M, M]
  const float* amp  = (const float*)d_in[4];  // scalar
  const float* ls   = (const float*)d_in[5];  // scalar
  float* out = (float*)d_out;                 // [N*U] sample.T, then [1] kl

  char* ws = (char*)d_ws;
  size_t off = 0;
  auto alloc = [&](size_t bytes) -> float* {
    float* p = (float*)(ws + off);
    off += (bytes + 255) & ~(size_t)255;
    return p;
  };
  float* nz     = alloc((size_t)UNITS * M * 4);
  float* nx     = alloc((size_t)N * 4);
  float* Kmm    = alloc((size_t)UNITS * M * M * 4);      // -> L in place
  float* Akmn   = alloc((size_t)UNITS * M * N * 4);      // Kmn -> A in place
  float* Cb     = alloc((size_t)UNITS * M * M * 4);
  float* alphab = alloc((size_t)UNITS * M * 4);
  float* Knn    = alloc((size_t)N * N * 4);
  float* CAb    = alloc((size_t)UNITS * M * N * 4);
  float* mub    = alloc((size_t)UNITS * N * 4);
  float* epsb   = alloc((size_t)UNITS * N * 4);
  float* covb   = alloc((size_t)UCHUNK * N * N * 4);

  row_norms_kernel<<<dim3((UNITS * M + 255) / 256), 256, 0, stream>>>(
      z, nz, UNITS * M);
  row_norms_kernel<<<dim3((N + 255) / 256), 256, 0, stream>>>(x, nx, N);

  // Kmm[u] = rbf(z,z) + JITTER*I  (tiles: 4x4 per unit)
  rbf_wmma_kernel<<<dim3(1, 4, UNITS), 128, 0, stream>>>(
      z, z, nz, nz, Kmm, (size_t)M * DD, (size_t)M * DD, M, M,
      (size_t)M * M, M, 4, amp, ls, JITTER);
  // Kmn[u] = rbf(z, x)  (tiles: 4x32 per unit)
  rbf_wmma_kernel<<<dim3(8, 4, UNITS), 128, 0, stream>>>(
      z, x, nz, nx, Akmn, (size_t)M * DD, 0, M, 0,
      (size_t)M * N, N, 32, amp, ls, 0.0f);
  // Knn = rbf(x, x)  (tiles: 32x32)
  rbf_wmma_kernel<<<dim3(8, 32, 1), 128, 0, stream>>>(
      x, x, nx, nx, Knn, 0, 0, 0, 0, 0, N, 32, amp, ls, 0.0f);

  cholesky64_kernel<<<UNITS, 64, 0, stream>>>(Kmm);
  trisolve_kernel<<<UNITS, 640, 0, stream>>>(Kmm, Akmn, Svar, Cb, loc, alphab);
  mu_kernel<<<UNITS, N, 0, stream>>>(Akmn, alphab, mub);
  ca_wmma_kernel<<<dim3(8, 4, UNITS), 128, 0, stream>>>(Cb, Akmn, CAb);
  eps_kernel<<<dim3((UNITS * N) / 256), 256, 0, stream>>>(epsb);

  kl_init_kernel<<<1, 32, 0, stream>>>(out + (size_t)N * UNITS);
  kl_kernel<<<UNITS, 256, 0, stream>>>(Cb, alphab, Kmm, Svar,
                                       out + (size_t)N * UNITS);

  for (int c = 0; c < UNITS / UCHUNK; ++c) {
    int ub = c * UCHUNK;
    cov_wmma_kernel<<<dim3(8, 32, UCHUNK), 128, 0, stream>>>(
        Akmn, CAb, Knn, covb, ub);
    cholesky512_kernel<<<UCHUNK, N, 0, stream>>>(covb);
    sample_kernel<<<dim3(2, UCHUNK), 256, 0, stream>>>(covb, mub, epsb, out, ub);
  }
}